// MultiHead_SelfAttention_88055419503376
// MI455X (gfx1250) — compile-verified
//
#include <hip/hip_runtime.h>
#include <hip/hip_bf16.h>
#include <math.h>

// CDNA5 / gfx1250. wave32.
typedef __attribute__((ext_vector_type(2))) float v2f;
typedef __attribute__((ext_vector_type(8))) float v8f;

// Problem constants from the reference: B=8, D=256, N=64 -> rows=2048, cols=192.
#define ROWS 2048   // B*D
#define KDIM 64     // N (contraction dim of projection)
#define COLS 192    // 3*N
#define NTN  (COLS / 16)   // 12 N-tiles
#define NTM  (ROWS / 16)   // 128 M-tiles

// ---------------------------------------------------------------------------
// Kernel 1: qkv = x(2048x64) @ W(64x192), exact fp32 via V_WMMA_F32_16X16X4_F32.
// One wave32 computes one 16x16 tile of qkv; K=64 -> 16 chained WMMAs.
//
// f32 WMMA operand layouts (ISA 7.12.2):
//   A (16x4, MxK):  lane l<16 : VGPR0=A[l][k0+0], VGPR1=A[l][k0+1]
//                   lane l>=16: VGPR0=A[l-16][k0+2], VGPR1=A[l-16][k0+3]
//   B (4x16, KxN):  symmetric (B row striped across lanes within a VGPR):
//                   lane l<16 : VGPR0=B[k0+0][l], VGPR1=B[k0+1][l]
//                   lane l>=16: VGPR0=B[k0+2][l-16], VGPR1=B[k0+3][l-16]
//   C/D (16x16):    VGPR r: lanes 0-15 -> D[r][n], lanes 16-31 -> D[r+8][n]
// ---------------------------------------------------------------------------
__global__ void __launch_bounds__(256)
qkv_gemm_wmma_f32(const float* __restrict__ x,
                  const float* __restrict__ W,
                  float* __restrict__ qkv) {
    const int lane  = threadIdx.x & 31;
    const int wave  = threadIdx.x >> 5;
    const int tile  = blockIdx.x * (blockDim.x >> 5) + wave;   // 0..1535
    if (tile >= NTM * NTN) return;                             // wave-uniform

    const int tm = tile / NTN;          // M tile (0..127)
    const int tn = tile % NTN;          // N tile (0..11)
    const int row0 = tm * 16;
    const int col0 = tn * 16;

    const int half = lane >> 4;         // 0: K pair {0,1}, 1: K pair {2,3}
    const int l    = lane & 15;         // row (A) / col (B) within tile

    // Exercise gfx1250 prefetch path (global_prefetch_b8).
    __builtin_prefetch(W + col0 + l, 0, 0);

    // Per-lane base pointers (k advances in the loop).
    const float* xr = x + (size_t)(row0 + l) * KDIM + 2 * half;     // A operand
    const float* wr = W + (size_t)(2 * half) * COLS + (col0 + l);   // B operand

    v8f c = {};
    #pragma unroll
    for (int k0 = 0; k0 < KDIM; k0 += 4) {
        v2f a, b;
        a.x = xr[k0 + 0];
        a.y = xr[k0 + 1];
        b.x = wr[(size_t)k0 * COLS];
        b.y = wr[(size_t)(k0 + 1) * COLS];
        // (neg_a, A, neg_b, B, c_mod, C, reuse_a, reuse_b)
        c = __builtin_amdgcn_wmma_f32_16x16x4_f32(
                false, a, false, b, (short)0, c, false, false);
    }

    // Scatter D: VGPR r -> row (row0 + r + 8*half), col (col0 + l).
    float* out = qkv + (size_t)(row0 + half * 8) * COLS + col0 + l;
    #pragma unroll
    for (int r = 0; r < 8; ++r)
        out[(size_t)r * COLS] = c[r];
}

// ---------------------------------------------------------------------------
// Kernel 2: per-(b, head) attention with head-dim 1.
//   scores[i,j] = q_i * k_j * scale ; softmax over j ; out_i = sum_j p_ij v_j
// Block = 256 threads = one (b, n). k/v columns staged in LDS.
// max_j(q_i*k_j) = max(q_i*kmax, q_i*kmin): the max pass collapses to a
// broadcast min/max scan of the k column.
// ---------------------------------------------------------------------------
__global__ void __launch_bounds__(256)
attn_headdim1(const float* __restrict__ qkv, float* __restrict__ out) {
    const int b = blockIdx.x >> 6;      // 0..7
    const int n = blockIdx.x & 63;      // 0..63
    const int t = threadIdx.x;          // 0..255 (= d index i)

    __shared__ float kcol[256];
    __shared__ float vcol[256];

    const float* base = qkv + (size_t)(b * 256) * COLS;
    kcol[t] = base[(size_t)t * COLS + KDIM + n];        // k[b, t, n]
    vcol[t] = base[(size_t)t * COLS + 2 * KDIM + n];    // v[b, t, n]
    __syncthreads();

    const float scale = 0.125f;                         // 1/sqrt(64)
    const float q = base[(size_t)t * COLS + n] * scale; // q[b, t, n] * scale

    // Block-broadcast reads: all lanes hit the same LDS word per iteration.
    float kmax = -INFINITY, kmin = INFINITY;
    for (int j = 0; j < 256; ++j) {
        const float kj = kcol[j];
        kmax = fmaxf(kmax, kj);
        kmin = fminf(kmin, kj);
    }
    const float m = fmaxf(q * kmax, q * kmin);          // row max of scores

    float s = 0.0f, o = 0.0f;
    for (int j = 0; j < 256; ++j) {
        const float e = __expf(fmaf(q, kcol[j], -m));   // v_exp_f32
        s += e;
        o = fmaf(e, vcol[j], o);
    }

    // out[b, t, n]
    out[((size_t)b * 256 + t) * KDIM + n] = o / s;
}

// ---------------------------------------------------------------------------
extern "C" void kernel_launch(void* const* d_in, const int* in_sizes, int n_in,
                              void* d_out, int out_size, void* d_ws, size_t ws_size,
                              hipStream_t stream) {
    const float* x = (const float*)d_in[0];   // (8, 256, 64)
    const float* W = (const float*)d_in[1];   // (64, 192)
    float* qkv = (float*)d_ws;                // (2048, 192) fp32 = 1.5 MB scratch
    float* out = (float*)d_out;               // (8, 256, 64)

    // 1536 tiles, 8 waves (256 thr) per block -> 192 blocks, no partial waves.
    qkv_gemm_wmma_f32<<<dim3(NTM * NTN / 8), dim3(256), 0, stream>>>(x, W, qkv);
    // One block per (b, head): 8*64 = 512 blocks.
    attn_headdim1<<<dim3(512), dim3(256), 0, stream>>>(qkv, out);
}